// RotatEModel_70866960384070
// MI455X (gfx1250) — compile-verified
//
#include <hip/hip_runtime.h>
#include <cstdint>

// ---------------------------------------------------------------------------
// RotatE batch-negative scoring for MI455X (gfx1250).
// Memory-bound gather (~1.07 GB of random 4KB rows vs ~46us HBM floor), so
// the optimization target is the data-movement path: double-buffered
// async global->LDS row staging (CDNA5 GLOBAL_LOAD_ASYNC_TO_LDS_B128 +
// S_WAIT_ASYNCCNT). WMMA is inapplicable: per-element sqrt breaks the
// bilinear structure, there is no matmul here.
// ---------------------------------------------------------------------------

#define EMB_D       500
#define TWO_D       1000
#define NUM_K       1024
#define NUM_B       256
#define MARGIN_F    9.0f
// PI / ((MARGIN+EPSILON)/EMB_DIM) = pi*500/11
#define PHASE_SCALE 142.79966607226332f

typedef __attribute__((ext_vector_type(4))) float v4f;
typedef __attribute__((ext_vector_type(4))) int   v4i;

#if defined(__has_builtin)
#  if __has_builtin(__builtin_amdgcn_global_load_async_to_lds_b128) && \
      __has_builtin(__builtin_amdgcn_s_wait_asynccnt)
#    define ROTATE_HAVE_ASYNC 1
#  endif
#endif
#ifndef ROTATE_HAVE_ASYNC
#  define ROTATE_HAVE_ASYNC 0
#endif

#if ROTATE_HAVE_ASYNC
// Builtin signature (from clang diagnostic): arg0 = int4 __device__* (as1),
// arg1 = int4 __shared__* (as3), then two int immediates (offset, cpol).
typedef __attribute__((address_space(1))) v4i* as1_v4i;
typedef __attribute__((address_space(3))) v4i* as3_v4i;

// Issue one 4000-byte entity row (1000 f32) as 8 per-wave async B128 copies.
// Chunk 7 is predicated (only 26 of 32 lanes) so we copy exactly 250 float4s
// and never read past the end of the 400MB entity table.
// ASYNCcnt increments by exactly 8 per call (one per instruction).
__device__ __forceinline__ void issue_row_async(const float* ent, int tail,
                                                float* buf, int lane) {
  const uint64_t g = (uint64_t)reinterpret_cast<uintptr_t>(ent)
                   + (uint64_t)(uint32_t)tail * 4000ull
                   + (uint32_t)(lane * 16);
  // Generic LDS pointer: low 32 bits are the wave-relative LDS byte offset.
  const uint32_t l = (uint32_t)reinterpret_cast<uintptr_t>(buf)
                   + (uint32_t)(lane * 16);
#pragma unroll
  for (int c = 0; c < 8; ++c) {
    if (c * 32 + lane < 250) {
      __builtin_amdgcn_global_load_async_to_lds_b128(
          (as1_v4i)(g + (uint32_t)(c * 512)),
          (as3_v4i)(l + (uint32_t)(c * 512)),
          /*offset=*/0, /*cpol=*/0);
    }
  }
}
#endif  // ROTATE_HAVE_ASYNC

__global__ void __launch_bounds__(128, 1)
rotate_score_kernel(const float* __restrict__ ent,
                    const float* __restrict__ rel,
                    const int*   __restrict__ h,
                    const int*   __restrict__ r,
                    const int*   __restrict__ t,
                    float*       __restrict__ out) {
  constexpr int WAVES = 4;                       // 128 threads, wave32
  constexpr int SPLIT = 16;                      // blocks per head b
  constexpr int KPW   = NUM_K / (SPLIT * WAVES); // 16 candidate rows per wave

  __shared__ __align__(16) float s_hr[TWO_D];    // rotated head: re[0..499], im[500..999]
#if ROTATE_HAVE_ASYNC
  __shared__ __align__(16) float s_buf[WAVES][2][1024];  // 2x 4KB per wave
#endif

  const int b    = blockIdx.x / SPLIT;
  const int part = blockIdx.x % SPLIT;

  // ---- Rotated head h*r into LDS (cheap; recomputed per block) -----------
  {
    const int hb = h[b], rb = r[b];
    const float* hrow = ent + (size_t)(uint32_t)hb * TWO_D;
    const float* rrow = rel + (size_t)(uint32_t)rb * EMB_D;
    for (int d = threadIdx.x; d < EMB_D; d += blockDim.x) {
      float s, c;
      sincosf(rrow[d] * PHASE_SCALE, &s, &c);
      const float reh = hrow[d];
      const float imh = hrow[EMB_D + d];
      s_hr[d]         = fmaf(reh, c, -imh * s);  // re_h*re_r - im_h*im_r
      s_hr[EMB_D + d] = fmaf(reh, s,  imh * c);  // re_h*im_r + im_h*re_r
    }
  }
  __syncthreads();

  const int wave = threadIdx.x >> 5;
  const int lane = threadIdx.x & 31;
  const int* trow = t + (size_t)b * NUM_K;
  const int kbase = part * (WAVES * KPW) + wave * KPW;

  // Per-lane partial: lanes cover float4 chunks 0..124 of the D=500 halves.
  auto row_partial = [&](const float* __restrict__ buf) -> float {
    float acc = 0.f;
#pragma unroll
    for (int it2 = 0; it2 < 4; ++it2) {
      const int m = it2 * 32 + lane;            // float4 index, 125 per half
      if (m < 125) {
        const v4f rt = *(const v4f*)(buf  +          4 * m);
        const v4f it = *(const v4f*)(buf  + EMB_D +  4 * m);
        const v4f rh = *(const v4f*)(s_hr +          4 * m);
        const v4f ih = *(const v4f*)(s_hr + EMB_D +  4 * m);
#pragma unroll
        for (int e = 0; e < 4; ++e) {
          const float dr = rh[e] - rt[e];
          const float di = ih[e] - it[e];
          acc += sqrtf(fmaf(dr, dr, di * di));
        }
      }
    }
    return acc;
  };

  auto finish = [&](float acc, int k) {
#pragma unroll
    for (int off = 16; off > 0; off >>= 1)
      acc += __shfl_xor(acc, off, 32);           // wave32 butterfly
    if (lane == 0) out[(size_t)b * NUM_K + k] = MARGIN_F - acc;
  };

#if ROTATE_HAVE_ASYNC
  // ---- Double-buffered async pipeline: row i lives in buf[i&1] -----------
  float* buf0 = &s_buf[wave][0][0];
  float* buf1 = &s_buf[wave][1][0];

  issue_row_async(ent, trow[kbase], buf0, lane);
#pragma unroll 1
  for (int i = 0; i < KPW - 1; ++i) {
    issue_row_async(ent, trow[kbase + i + 1], (i & 1) ? buf0 : buf1, lane);
    __builtin_amdgcn_s_wait_asynccnt(8);         // row i's 8 copies complete
    finish(row_partial((i & 1) ? buf1 : buf0), kbase + i);
  }
  __builtin_amdgcn_s_wait_asynccnt(0);
  finish(row_partial(buf1), kbase + KPW - 1);    // (KPW-1)&1 == 1
#else
  // ---- Fallback: direct coalesced float4 global loads ---------------------
#pragma unroll 1
  for (int i = 0; i < KPW; ++i) {
    const float* row = ent + (size_t)(uint32_t)trow[kbase + i] * TWO_D;
    finish(row_partial(row), kbase + i);
  }
#endif
}

extern "C" void kernel_launch(void* const* d_in, const int* in_sizes, int n_in,
                              void* d_out, int out_size, void* d_ws, size_t ws_size,
                              hipStream_t stream) {
  (void)in_sizes; (void)n_in; (void)out_size; (void)d_ws; (void)ws_size;
  const float* ent = (const float*)d_in[0];
  const float* rel = (const float*)d_in[1];
  const int*   h   = (const int*)d_in[2];
  const int*   r   = (const int*)d_in[3];
  const int*   t   = (const int*)d_in[4];
  float*       out = (float*)d_out;

  dim3 grid(NUM_B * 16);   // 4096 blocks: 16 per head b
  dim3 block(128);         // 4 wave32 waves
  hipLaunchKernelGGL(rotate_score_kernel, grid, block, 0, stream,
                     ent, rel, h, r, t, out);
}